// GAT_83296595739028
// MI455X (gfx1250) — compile-verified
//
#include <hip/hip_runtime.h>

#define ALPHA    0.2f
#define NHEADS   3
#define NFEAT    256
#define NHID     64
#define NH2      192   // NHEADS*NHID, layer-2 K
#define NC       40    // NCLASS
#define NCP      48    // NC padded to 3 WMMA tiles

typedef __attribute__((ext_vector_type(2))) float v2f;
typedef __attribute__((ext_vector_type(8))) float v8f;

// ---- helpers -------------------------------------------------------------

__device__ __forceinline__ v8f wmma_f32(v2f a, v2f b, v8f c) {
    // V_WMMA_F32_16X16X4_F32 : D = A(16x4) * B(4x16) + C(16x16), fp32
    return __builtin_amdgcn_wmma_f32_16x16x4_f32(
        /*neg_a=*/false, a, /*neg_b=*/false, b,
        /*c_mod=*/(short)0, c, /*reuse_a=*/false, /*reuse_b=*/false);
}

__device__ __forceinline__ float elu_f(float v) {
    return v > 0.0f ? v : (__expf(v) - 1.0f);
}

__device__ __forceinline__ float edge_w(float logit) {
    float lr = logit > 0.0f ? logit : ALPHA * logit;   // leaky_relu
    return __expf(-lr);                                 // exp(-leakyrelu)
}

// ---- 0: zero accumulators ------------------------------------------------

__global__ void zero_kernel(float4* __restrict__ p, long n4) {
    long i = (long)blockIdx.x * blockDim.x + threadIdx.x;
    long stride = (long)gridDim.x * blockDim.x;
    float4 z = {0.f, 0.f, 0.f, 0.f};
    for (; i < n4; i += stride) p[i] = z;
}

// ---- 1: GEMM layer 1: h1[n][head*64+f] = x[n][:] @ W[head][:, f] ---------
// grid (ceil(N/64), NHEADS), block 128 (4 waves). Each wave: 16 rows x 64 cols.
// LDS holds a 128-K stage of W in PAIR-INTERLEAVED layout:
//   lw[(k/2)*128 + col*2 + (k&1)] = W[ks+k][col]
// so a B fragment {W[kb][c], W[kb+1][c]} is one aligned ds_load_b64.

__global__ __launch_bounds__(128) void gemm1_kernel(
    const float* __restrict__ x, const float* __restrict__ W,
    float* __restrict__ h1, int N)
{
    __shared__ float lw[128 * NHID];           // 32 KB stage
    const int head = blockIdx.y;
    const float* Wh = W + (size_t)head * NFEAT * NHID;

    const int wave = threadIdx.x >> 5;
    const int lane = threadIdx.x & 31;
    const int half = lane >> 4;                // K sub-pair selector
    const int l16  = lane & 15;
    const int m0   = blockIdx.x * 64 + wave * 16;

    int row = m0 + l16;
    if (row >= N) row = N - 1;                 // clamp; stores are guarded
    const float* xr = x + (size_t)row * NFEAT + half * 2;

    v8f acc0 = {}, acc1 = {}, acc2 = {}, acc3 = {};

    for (int ks = 0; ks < NFEAT; ks += 128) {
        __syncthreads();
        // cooperative pair-interleaved fill: 128 rows x 64 cols
        for (int i = threadIdx.x; i < 128 * NHID / 2; i += 128) {
            int r  = i >> 5;                   // row in stage (NHID/2 f2 per row)
            int c2 = (i & 31) * 2;             // starting column of the pair
            float2 wv = ((const float2*)(Wh + (size_t)(ks + r) * NHID))[i & 31];
            float* dp = lw + (r >> 1) * (2 * NHID) + (r & 1);
            dp[c2 * 2]       = wv.x;
            dp[(c2 + 1) * 2] = wv.y;
        }
        __syncthreads();

        const float* bb = lw + half * (2 * NHID) + l16 * 2;   // per-lane B base
        v2f a = *(const v2f*)(xr + ks);                        // step kk=0
        #pragma unroll
        for (int kk = 0; kk < 128; kk += 4) {
            v2f an = a;
            if (kk < 124) an = *(const v2f*)(xr + ks + kk + 4); // pipeline A
            const float* bs = bb + kk * (NHID / 2) * 2;         // (kk/2)*2*NHID
            v2f b0 = *(const v2f*)(bs);
            v2f b1 = *(const v2f*)(bs + 32);
            v2f b2 = *(const v2f*)(bs + 64);
            v2f b3 = *(const v2f*)(bs + 96);
            acc0 = wmma_f32(a, b0, acc0);
            acc1 = wmma_f32(a, b1, acc1);
            acc2 = wmma_f32(a, b2, acc2);
            acc3 = wmma_f32(a, b3, acc3);
            a = an;
        }
    }

    // C/D layout: VGPR r -> row (m0 + r + 8*half), col n0 + l16
    for (int r = 0; r < 8; ++r) {
        int orow = m0 + r + half * 8;
        if (orow < N) {
            float* op = h1 + (size_t)orow * NH2 + head * NHID;
            op[l16]      = acc0[r];
            op[16 + l16] = acc1[r];
            op[32 + l16] = acc2[r];
            op[48 + l16] = acc3[r];
        }
    }
}

// ---- 2: per-(node,head) attention dots: ad = h.a1, as = h.a2 -------------

__global__ __launch_bounds__(256) void dots1_kernel(
    const float* __restrict__ h1, const float* __restrict__ a1,
    const float* __restrict__ a2, float* __restrict__ ad,
    float* __restrict__ as, int N)
{
    int wid  = (int)(((long)blockIdx.x * blockDim.x + threadIdx.x) >> 5);
    int lane = threadIdx.x & 31;
    if (wid >= NHEADS * N) return;
    int k = wid / N;
    int n = wid - k * N;

    const float* hp  = h1 + (size_t)n * NH2 + k * NHID;
    const float* a1p = a1 + k * NHID;
    const float* a2p = a2 + k * NHID;
    float h0 = hp[lane], h1v = hp[lane + 32];
    float pd = h0 * a1p[lane] + h1v * a1p[lane + 32];
    float ps = h0 * a2p[lane] + h1v * a2p[lane + 32];
    for (int off = 16; off > 0; off >>= 1) {
        pd += __shfl_down(pd, off);
        ps += __shfl_down(ps, off);
    }
    if (lane == 0) { ad[wid] = pd; as[wid] = ps; }
}

// ---- 3: layer-1 edge scatter (wave per edge) -----------------------------

__global__ __launch_bounds__(256) void edge1_kernel(
    const int* __restrict__ ei, long E, int N,
    const float* __restrict__ h1, const float* __restrict__ ad,
    const float* __restrict__ as, float* __restrict__ denom,
    float* __restrict__ num)
{
    long e   = (((long)blockIdx.x * blockDim.x + threadIdx.x) >> 5);
    int lane = threadIdx.x & 31;
    if (e >= E) return;
    int dst = ei[e];
    int src = ei[E + e];

    #pragma unroll
    for (int k = 0; k < NHEADS; ++k) {
        float w = edge_w(ad[(size_t)k * N + dst] + as[(size_t)k * N + src]);
        if (lane == 0) atomicAdd(&denom[(size_t)k * N + dst], w);
        const float* hs = h1  + (size_t)src * NH2 + k * NHID;
        float*       np = num + (size_t)dst * NH2 + k * NHID;
        atomicAdd(&np[lane],      w * hs[lane]);
        atomicAdd(&np[lane + 32], w * hs[lane + 32]);
    }
}

// ---- 4: normalize + ELU in place (num1 becomes h_cat) --------------------

__global__ void norm1_kernel(float* __restrict__ num,
                             const float* __restrict__ denom, int N)
{
    long total = (long)N * NH2;
    long i = (long)blockIdx.x * blockDim.x + threadIdx.x;
    long stride = (long)gridDim.x * blockDim.x;
    for (; i < total; i += stride) {
        long n = i / NH2;
        int  c = (int)(i - n * NH2);
        int  k = c >> 6;
        float v = num[i] / (denom[(size_t)k * N + n] + 1e-16f);
        num[i] = elu_f(v);
    }
}

// ---- 5: GEMM layer 2: h2 = h_cat[N,192] @ W_out[192,40] ------------------
// Same pair-interleaved LDS trick; 48-col padded, row-pair length 96 floats.

__global__ __launch_bounds__(128) void gemm2_kernel(
    const float* __restrict__ hcat, const float* __restrict__ W2,
    float* __restrict__ h2, int N)
{
    __shared__ float lw[(NH2 / 2) * (2 * NCP)];   // 96 pairs * 96 floats = 36 KB
    for (int i = threadIdx.x; i < NH2 * NCP; i += 128) {
        int r = i / NCP, c = i - r * NCP;
        float v = (c < NC) ? W2[r * NC + c] : 0.0f;
        lw[(r >> 1) * (2 * NCP) + c * 2 + (r & 1)] = v;
    }
    __syncthreads();

    const int wave = threadIdx.x >> 5;
    const int lane = threadIdx.x & 31;
    const int half = lane >> 4;
    const int l16  = lane & 15;
    const int m0   = blockIdx.x * 64 + wave * 16;

    int row = m0 + l16;
    if (row >= N) row = N - 1;
    const float* xr = hcat + (size_t)row * NH2 + half * 2;

    const float* bb = lw + half * (2 * NCP) + l16 * 2;
    v8f acc0 = {}, acc1 = {}, acc2 = {};
    v2f a = *(const v2f*)(xr);
    #pragma unroll
    for (int k = 0; k < NH2; k += 4) {
        v2f an = a;
        if (k < NH2 - 4) an = *(const v2f*)(xr + k + 4);       // pipeline A
        const float* bs = bb + (k >> 1) * (2 * NCP);
        v2f b0 = *(const v2f*)(bs);
        v2f b1 = *(const v2f*)(bs + 32);
        v2f b2 = *(const v2f*)(bs + 64);
        acc0 = wmma_f32(a, b0, acc0);
        acc1 = wmma_f32(a, b1, acc1);
        acc2 = wmma_f32(a, b2, acc2);
        a = an;
    }

    for (int r = 0; r < 8; ++r) {
        int orow = m0 + r + half * 8;
        if (orow < N) {
            float* op = h2 + (size_t)orow * NC;
            op[l16]      = acc0[r];
            op[16 + l16] = acc1[r];
            if (l16 < 8) op[32 + l16] = acc2[r];   // cols 40..47 are padding
        }
    }
}

// ---- 6: layer-2 dots -----------------------------------------------------

__global__ __launch_bounds__(256) void dots2_kernel(
    const float* __restrict__ h2, const float* __restrict__ a1o,
    const float* __restrict__ a2o, float* __restrict__ ad,
    float* __restrict__ as, int N)
{
    int wid  = (int)(((long)blockIdx.x * blockDim.x + threadIdx.x) >> 5);
    int lane = threadIdx.x & 31;
    if (wid >= N) return;
    const float* hp = h2 + (size_t)wid * NC;
    float pd = hp[lane] * a1o[lane];
    float ps = hp[lane] * a2o[lane];
    if (lane < NC - 32) {
        pd += hp[lane + 32] * a1o[lane + 32];
        ps += hp[lane + 32] * a2o[lane + 32];
    }
    for (int off = 16; off > 0; off >>= 1) {
        pd += __shfl_down(pd, off);
        ps += __shfl_down(ps, off);
    }
    if (lane == 0) { ad[wid] = pd; as[wid] = ps; }
}

// ---- 7: layer-2 edge scatter ---------------------------------------------

__global__ __launch_bounds__(256) void edge2_kernel(
    const int* __restrict__ ei, long E,
    const float* __restrict__ h2, const float* __restrict__ ad,
    const float* __restrict__ as, float* __restrict__ denom,
    float* __restrict__ num)
{
    long e   = (((long)blockIdx.x * blockDim.x + threadIdx.x) >> 5);
    int lane = threadIdx.x & 31;
    if (e >= E) return;
    int dst = ei[e];
    int src = ei[E + e];
    float w = edge_w(ad[dst] + as[src]);
    if (lane == 0) atomicAdd(&denom[dst], w);
    const float* hs = h2  + (size_t)src * NC;
    float*       np = num + (size_t)dst * NC;
    atomicAdd(&np[lane], w * hs[lane]);
    if (lane < NC - 32) atomicAdd(&np[lane + 32], w * hs[lane + 32]);
}

// ---- 8: final normalize + ELU --------------------------------------------

__global__ void final_kernel(const float* __restrict__ num,
                             const float* __restrict__ denom,
                             float* __restrict__ out, int N)
{
    long total = (long)N * NC;
    long i = (long)blockIdx.x * blockDim.x + threadIdx.x;
    long stride = (long)gridDim.x * blockDim.x;
    for (; i < total; i += stride) {
        long n = i / NC;
        float v = num[i] / (denom[n] + 1e-16f);
        out[i] = elu_f(v);
    }
}

// ---- host launcher -------------------------------------------------------

extern "C" void kernel_launch(void* const* d_in, const int* in_sizes, int n_in,
                              void* d_out, int out_size, void* d_ws, size_t ws_size,
                              hipStream_t stream) {
    const float* x    = (const float*)d_in[0];
    const int*   ei   = (const int*)d_in[1];          // [2,E] (jax default int32)
    const float* W    = (const float*)d_in[2];        // [3,256,64]
    const float* a1   = (const float*)d_in[3];        // [3,64]
    const float* a2   = (const float*)d_in[4];        // [3,64]
    const float* W2   = (const float*)d_in[5];        // [192,40]
    const float* a1o  = (const float*)d_in[6];        // [40]
    const float* a2o  = (const float*)d_in[7];        // [40]
    float* out = (float*)d_out;

    const int  N = in_sizes[0] / NFEAT;               // 100000
    const long E = (long)in_sizes[1] / 2;             // 3300000

    // workspace layout (floats); accumulators contiguous at the end for one memset
    float* ws      = (float*)d_ws;
    float* h1      = ws;                              // 192N
    float* ad1     = h1   + (size_t)192 * N;          // 3N
    float* as1     = ad1  + (size_t)3 * N;            // 3N
    float* h2      = as1  + (size_t)3 * N;            // 40N
    float* ad2     = h2   + (size_t)40 * N;           // N
    float* as2     = ad2  + (size_t)N;                // N
    float* den1    = as2  + (size_t)N;                // 3N   -- ACC region start
    float* num1    = den1 + (size_t)3 * N;            // 192N (becomes h_cat)
    float* den2    = num1 + (size_t)192 * N;          // N
    float* num2    = den2 + (size_t)N;                // 40N  -- ACC region end
    const long acc_floats = (long)(3 + 192 + 1 + 40) * N;   // 236N, /4 exact

    // 0: zero accumulators
    {
        long n4 = acc_floats / 4;
        int blocks = (int)((n4 + 255) / 256);
        if (blocks > 65535) blocks = 65535;
        zero_kernel<<<blocks, 256, 0, stream>>>((float4*)den1, n4);
    }
    // 1: layer-1 GEMM (WMMA fp32)
    {
        dim3 grid((N + 63) / 64, NHEADS);
        gemm1_kernel<<<grid, 128, 0, stream>>>(x, W, h1, N);
    }
    // 2: layer-1 attention dots
    {
        long waves = (long)NHEADS * N;
        dots1_kernel<<<(int)((waves + 7) / 8), 256, 0, stream>>>(h1, a1, a2, ad1, as1, N);
    }
    // 3: layer-1 edge scatter
    edge1_kernel<<<(int)((E + 7) / 8), 256, 0, stream>>>(ei, E, N, h1, ad1, as1, den1, num1);
    // 4: normalize + ELU -> h_cat (in place in num1)
    {
        long total = (long)N * NH2;
        int blocks = (int)((total + 255) / 256);
        if (blocks > 131072) blocks = 131072;
        norm1_kernel<<<blocks, 256, 0, stream>>>(num1, den1, N);
    }
    // 5: layer-2 GEMM (WMMA fp32)
    gemm2_kernel<<<(N + 63) / 64, 128, 0, stream>>>(num1, W2, h2, N);
    // 6: layer-2 dots
    dots2_kernel<<<(N + 7) / 8, 256, 0, stream>>>(h2, a1o, a2o, ad2, as2, N);
    // 7: layer-2 edge scatter
    edge2_kernel<<<(int)((E + 7) / 8), 256, 0, stream>>>(ei, E, h2, ad2, as2, den2, num2);
    // 8: final normalize + ELU
    {
        long total = (long)N * NC;
        int blocks = (int)((total + 255) / 256);
        if (blocks > 131072) blocks = 131072;
        final_kernel<<<blocks, 256, 0, stream>>>(num2, den2, out, N);
    }
}